// DoubleEKVBlock_12154757447735
// MI455X (gfx1250) — compile-verified
//
#include <hip/hip_runtime.h>

// ---------------------------------------------------------------------------
// DoubleEKVBlock for MI455X (gfx1250).
// Math: per output (n,o,h,w):
//   v = ALPHA*TIA * sum_p [ F((z-thp)/PHI) - F((z-thn)/PHI) ]
//   F(u) = softplus(u)^2 - softplus(u-DELTA)^2,  DELTA = VD/PHI
// Transcendental-bound (no bilinear structure -> WMMA not applicable).
// CDNA5 path used: TDM tensor_load_to_lds for LDS staging (+ LDS padding
// feature to build 34-wide halo rows), s_wait_tensorcnt, wave-uniform LDS
// theta broadcasts.
// ---------------------------------------------------------------------------

#define INV_PHI   12.820512820512821f   // 1/0.078
#define DELTA_F   2.5641025641025643f   // 0.2/0.078
#define CEXP_F    0.0769882f            // exp(-DELTA)
#define ALPHA_TIA 0.02f                 // 1e-5 * 2000

typedef unsigned int v4u __attribute__((ext_vector_type(4)));
typedef int          v4i __attribute__((ext_vector_type(4)));
typedef int          v8i __attribute__((ext_vector_type(8)));

__device__ __forceinline__ unsigned lds_off(const void* p) {
  // LDS aperture: generic addr [31:0] is the workgroup-relative byte offset.
  return (unsigned)(unsigned long long)p;
}

// TDM load: up-to-2D tile (tileX contiguous f32, tileY lines at strideElems),
// optional LDS pad control bits (pad_enable/interval/amount) in padCtl.
__device__ __forceinline__ void tdm_load(unsigned ldsOff, const float* gptr,
                                         unsigned tileX, unsigned tileY,
                                         unsigned strideElems, unsigned padCtl) {
  unsigned long long ga = (unsigned long long)(const void*)gptr;
  v4u g0;
  g0[0] = 1u;                                           // count=1, user mode
  g0[1] = ldsOff;                                       // lds_addr (bytes)
  g0[2] = (unsigned)ga;                                 // global_addr[31:0]
  g0[3] = (unsigned)((ga >> 32) & 0x1FFFFFFull)         // global_addr[56:32]
        | (2u << 30);                                   // type=2 ("image")
  v8i g1;
  g1[0] = (int)((2u << 16) | padCtl);                   // data_size=4B + pad ctl
  g1[1] = (int)((tileX & 0xFFFFu) << 16);               // tensor_dim0 lo16
  g1[2] = (int)(((tileX >> 16) & 0xFFFFu)               // tensor_dim0 hi16
        | ((tileY & 0xFFFFu) << 16));                   // tensor_dim1 lo16
  g1[3] = (int)(((tileY >> 16) & 0xFFFFu)               // tensor_dim1 hi16
        | ((tileX & 0xFFFFu) << 16));                   // tile_dim0
  g1[4] = (int)(tileY & 0xFFFFu);                       // tile_dim1 (tile_dim2=0)
  g1[5] = (int)strideElems;                             // tensor_dim0_stride lo32
  g1[6] = 0;
  g1[7] = 0;
  v4i g2, g3;
  g2[0] = 1; g2[1] = 1; g2[2] = 0; g2[3] = 0;           // tensor_dim2/3 = 1 (benign)
  g3[0] = 0; g3[1] = (int)(1u << 16); g3[2] = 0; g3[3] = 0; // tensor_dim4 = 1
  v8i g4 = {0, 0, 0, 0, 0, 0, 0, 0};                    // trailing group (unused)
  __builtin_amdgcn_tensor_load_to_lds(g0, g1, g2, g3, g4, 0);
}

// F(u) = sp(u)^2 - sp(u-d)^2 with one shared exp; exact linear tail for big u.
__device__ __forceinline__ float Fdiff(float u) {
  float e   = __expf(fminf(u, 33.0f));          // e<=2e14, never overflows
  float sp1 = (u > 30.0f) ? u : __logf(1.0f + e);
  float um  = u - DELTA_F;
  float sp2 = (um > 30.0f) ? um : __logf(fmaf(CEXP_F, e, 1.0f));
  return (sp1 - sp2) * (sp1 + sp2);
}

// ---------------------------------------------------------------------------
// Kernel 1: voltage mapper (elementwise, per-channel scale/bias).
// ---------------------------------------------------------------------------
__global__ void ekv_vm_kernel(const float* __restrict__ x,
                              const float* __restrict__ scale,
                              const float* __restrict__ bias,
                              float* __restrict__ z, int total) {
  int i = blockIdx.x * blockDim.x + threadIdx.x;
  if (i >= total) return;
  int c = (i >> 10) & 31;                               // (i / (32*32)) % 32
  float rs = fminf(fmaxf(fabsf(scale[c]), 0.01f), 3.0f);
  z[i] = fminf(fmaxf(fmaf(x[i], rs, bias[c]), 0.0f), 8.0f);
}

// ---------------------------------------------------------------------------
// Conv kernel. Grid: 512 blocks = n(4) * h(32) * oQuarter(4). 256 threads:
// wave (0..7) -> o = oq*8+wave, lane (0..31) -> w. One lane per output elem.
// EPI=0: epilogue = voltage mapper 2 (writes z2). EPI=1: residual ReLU.
// ---------------------------------------------------------------------------
template <int EPI>
__global__ __launch_bounds__(256)
void ekv_conv_kernel(const float* __restrict__ zin,
                     const float* __restrict__ thp,
                     const float* __restrict__ thn,
                     const float* __restrict__ a1,   // scale2 (EPI=0) or x (EPI=1)
                     const float* __restrict__ a2,   // bias2  (EPI=0) or unused
                     float* __restrict__ out) {
  __shared__ float smem[3264 + 2 * 2304];   // ztile[3][32][34] + thp/thn slices
  float* zt  = smem;
  float* stp = smem + 3264;
  float* stn = stp + 2304;

  const int tid  = threadIdx.x;
  const int b    = blockIdx.x;
  const int oq   = b & 3;
  const int h    = (b >> 2) & 31;
  const int n    = b >> 7;
  const int wave = tid >> 5;
  const int w    = tid & 31;
  const int o    = oq * 8 + wave;

  // Zero the halo tile: covers unfold's literal-0 padding (cols 0/33, OOB rows)
  for (int i = tid; i < 3264; i += 256) zt[i] = 0.0f;
  __syncthreads();

  if (tid == 0) {
    // theta slices for this o-quarter: 8*288 contiguous floats each (1D TDM)
    tdm_load(lds_off(stp), thp + (size_t)(oq * 8) * 288, 2304, 1, 2304, 0);
    tdm_load(lds_off(stn), thn + (size_t)(oq * 8) * 288, 2304, 1, 2304, 0);
    // z halo rows: 2D tile 32(w) x 32(c), channel stride H*W=1024; TDM LDS
    // padding inserts 2 DWORDs after every 32 -> 34-col rows, lands at col 1.
    const unsigned padCtl = (1u << 20) | (4u << 22) | (1u << 25);
    for (int r = 0; r < 3; ++r) {
      int row = h - 1 + r;
      if (row >= 0 && row < 32)
        tdm_load(lds_off(zt + r * 1088 + 1),
                 zin + (size_t)n * 32768 + (size_t)row * 32,
                 32, 32, 1024, padCtl);
    }
    __builtin_amdgcn_s_wait_tensorcnt(0);
  }
  __syncthreads();

  // Prescale thetas by 1/PHI once (saves a VALU op per eval in the hot loop)
  for (int i = tid; i < 2304; i += 256) {
    stp[i] *= INV_PHI;
    stn[i] *= INV_PHI;
  }
  __syncthreads();

  const float* tpo = stp + wave * 288;   // wave-uniform -> LDS broadcast reads
  const float* tno = stn + wave * 288;

  float acc = 0.0f;
  for (int c = 0; c < 32; ++c) {
#pragma unroll
    for (int r = 0; r < 3; ++r) {
      const float* zrow = zt + r * 1088 + c * 34 + w;  // lanes read consecutive
      const int pb = c * 9 + r * 3;
#pragma unroll
      for (int kw = 0; kw < 3; ++kw) {
        float zi = zrow[kw] * INV_PHI;
        acc += Fdiff(zi - tpo[pb + kw]);
        acc -= Fdiff(zi - tno[pb + kw]);
      }
    }
  }

  float v = acc * ALPHA_TIA;
  size_t oidx = (size_t)((n * 32 + o) * 32 + h) * 32 + w;
  if (EPI == 0) {
    float rs = fminf(fmaxf(fabsf(a1[o]), 0.01f), 3.0f);
    out[oidx] = fminf(fmaxf(fmaf(v, rs, a2[o]), 0.0f), 8.0f);
  } else {
    out[oidx] = fmaxf(v + a1[oidx], 0.0f);
  }
}

// ---------------------------------------------------------------------------
extern "C" void kernel_launch(void* const* d_in, const int* in_sizes, int n_in,
                              void* d_out, int out_size, void* d_ws, size_t ws_size,
                              hipStream_t stream) {
  const float* x   = (const float*)d_in[0];
  const float* s1  = (const float*)d_in[1];
  const float* b1  = (const float*)d_in[2];
  const float* tp1 = (const float*)d_in[3];
  const float* tn1 = (const float*)d_in[4];
  const float* s2  = (const float*)d_in[5];
  const float* b2  = (const float*)d_in[6];
  const float* tp2 = (const float*)d_in[7];
  const float* tn2 = (const float*)d_in[8];
  float* outp = (float*)d_out;

  float* z1 = (float*)d_ws;            // N*C*H*W = 131072 floats
  float* z2 = z1 + 131072;             // N*O*H*W = 131072 floats

  ekv_vm_kernel<<<512, 256, 0, stream>>>(x, s1, b1, z1, 131072);
  ekv_conv_kernel<0><<<512, 256, 0, stream>>>(z1, tp1, tn1, s2, b2, z2);
  ekv_conv_kernel<1><<<512, 256, 0, stream>>>(z2, tp2, tn2, x, nullptr, outp);
}